// Attention_13589276525011
// MI455X (gfx1250) — compile-verified
//
#include <hip/hip_runtime.h>
#include <math.h>

#define B_      64
#define T_      2048
#define RNN_DIM 1024
#define EMB_DIM 512
#define ATT_DIM 128
#define N_FILT  32
#define KSIZE   31
#define PAD_    15
#define TT      64              // t-tile per workgroup in k1

typedef __attribute__((ext_vector_type(2))) float v2f;
typedef __attribute__((ext_vector_type(8))) float v8f;

// ---------------------------------------------------------------------------
// K0: pq[b][a] = dot(hidden[b, :1024], Wq[a, :1024])    (64 x 128, tiny)
// ---------------------------------------------------------------------------
__global__ void k0_pq(const float* __restrict__ hidden,   // (B, RNN_DIM)
                      const float* __restrict__ Wq,       // (ATT_DIM, RNN_DIM)
                      float* __restrict__ pq)             // (B, ATT_DIM)
{
    const int b = blockIdx.x;
    const int a = threadIdx.x;                 // 128 threads
    const float* __restrict__ h = hidden + (size_t)b * RNN_DIM;
    const float* __restrict__ w = Wq + (size_t)a * RNN_DIM;
    float acc = 0.f;
#pragma unroll 4
    for (int i = 0; i < RNN_DIM; ++i)
        acc = fmaf(h[i], w[i], acc);
    pq[b * ATT_DIM + a] = acc;
}

// ---------------------------------------------------------------------------
// K1: fused conv1d + (loc @ Wp^T via V_WMMA_F32_16X16X4_F32) + tanh-dot
//     energies[b, t] = sum_a Wv[a] * tanh(pq[b,a] + proj[t,a] + pm[b,t,a])
// One 256-thread workgroup per (t-tile of 64, batch row).
// ---------------------------------------------------------------------------
__global__ void __launch_bounds__(256)
k1_energies(const float* __restrict__ cat,    // (B, 2, T)
            const float* __restrict__ pm,     // (B, T, ATT_DIM)
            const float* __restrict__ convw,  // (N_FILT, 2, KSIZE)
            const float* __restrict__ Wp,     // (ATT_DIM, N_FILT)
            const float* __restrict__ Wv,     // (1, ATT_DIM)
            const float* __restrict__ pq,     // (B, ATT_DIM)
            float* __restrict__ energies)     // (B, T)
{
    __shared__ float sCat[2][TT + KSIZE + 1];       // 2 x 96 (padded slice)
    __shared__ float sW[N_FILT * 2 * KSIZE];        // conv weights, 1984
    __shared__ float sLoc[TT][N_FILT + 1];          // conv out tile (t x f)
    __shared__ float sProj[TT][ATT_DIM + 4];        // projected tile (t x a)
    __shared__ float sPq[ATT_DIM];
    __shared__ float sWv[ATT_DIM];
    __shared__ float sE[TT];

    const int tid  = threadIdx.x;                   // 256 threads = 8 waves
    const int b    = blockIdx.y;
    const int t0   = blockIdx.x * TT;
    const int lane = tid & 31;
    const int wave = tid >> 5;

    // ---- stage 0: cooperative loads -------------------------------------
    for (int i = tid; i < 2 * (TT + KSIZE - 1); i += 256) {
        const int c = i / (TT + KSIZE - 1);
        const int x = i % (TT + KSIZE - 1);
        const int tg = t0 + x - PAD_;
        sCat[c][x] = (tg >= 0 && tg < T_) ? cat[((size_t)b * 2 + c) * T_ + tg] : 0.f;
    }
    for (int i = tid; i < N_FILT * 2 * KSIZE; i += 256)
        sW[i] = convw[i];
    if (tid < ATT_DIM) {
        sPq[tid] = pq[b * ATT_DIM + tid];
        sWv[tid] = Wv[tid];
    }
    if (tid < TT) sE[tid] = 0.f;
    __syncthreads();

    // ---- stage 1: conv1d -> sLoc[t][f] ----------------------------------
    {
        const int f  = tid & 31;
        const int tb = tid >> 5;                    // 8 t-rows each
        const float* __restrict__ w0 = &sW[f * 2 * KSIZE];
#pragma unroll
        for (int j = 0; j < TT / 8; ++j) {
            const int t = tb * (TT / 8) + j;
            float acc = 0.f;
#pragma unroll
            for (int k = 0; k < KSIZE; ++k) {
                acc = fmaf(sCat[0][t + k], w0[k], acc);
                acc = fmaf(sCat[1][t + k], w0[KSIZE + k], acc);
            }
            sLoc[t][f] = acc;
        }
    }
    __syncthreads();

    // ---- stage 2: proj(64x128) = loc(64x32) @ Wp^T(32x128) via WMMA -----
    // Wave `wave` owns N-tile n0 = 16*wave, iterates 4 M-tiles, 8 K-steps.
    {
        const int n0 = wave * 16;
        const int ln = lane & 15;
        const int kh = (lane >> 4) << 1;            // 0 (lanes 0-15) or 2

        // B fragments (4x16, B[k][n] = Wp[n0+n][k]); reused for all M-tiles
        v2f bfrag[8];
#pragma unroll
        for (int ks = 0; ks < 8; ++ks) {
            const int k0 = ks * 4;
            bfrag[ks].x = Wp[(n0 + ln) * N_FILT + k0 + kh + 0];
            bfrag[ks].y = Wp[(n0 + ln) * N_FILT + k0 + kh + 1];
        }

#pragma unroll
        for (int mt = 0; mt < 4; ++mt) {
            const int m0 = mt * 16;
            v8f acc = {};
#pragma unroll
            for (int ks = 0; ks < 8; ++ks) {
                const int k0 = ks * 4;
                v2f af;
                af.x = sLoc[m0 + ln][k0 + kh + 0];
                af.y = sLoc[m0 + ln][k0 + kh + 1];
                acc = __builtin_amdgcn_wmma_f32_16x16x4_f32(
                    /*neg_a=*/false, af, /*neg_b=*/false, bfrag[ks],
                    /*c_mod=*/(short)0, acc, /*reuse_a=*/false, /*reuse_b=*/false);
            }
            // D layout: VGPR r -> (M = m0 + r + (lane<16 ? 0 : 8), N = lane&15)
            const int mh = (lane >> 4) * 8;
#pragma unroll
            for (int r = 0; r < 8; ++r)
                sProj[m0 + mh + r][n0 + ln] = acc[r];
        }
    }
    __syncthreads();

    // ---- stage 3: energies with streaming pm ----------------------------
    {
        const int a  = tid & 127;                   // ATT lane
        const int tl = tid >> 7;                    // 0/1: two t rows in flight
        const float wv  = sWv[a];
        const float pqa = sPq[a];
        const float* __restrict__ pmb = pm + ((size_t)b * T_ + t0) * ATT_DIM;
        for (int t = tl; t < TT; t += 2) {
            const float m = __builtin_nontemporal_load(&pmb[(size_t)t * ATT_DIM + a]);
            float v = tanhf(pqa + sProj[t][a] + m) * wv;
#pragma unroll
            for (int off = 16; off > 0; off >>= 1)
                v += __shfl_down(v, off, 32);
            if (lane == 0) atomicAdd(&sE[t], v);    // ds_add_f32
        }
    }
    __syncthreads();
    if (tid < TT)
        energies[(size_t)b * T_ + t0 + tid] = sE[tid];
}

// ---------------------------------------------------------------------------
// K2: softmax over T per batch row; writes attention_weights to output.
// ---------------------------------------------------------------------------
__global__ void __launch_bounds__(256)
k2_softmax(const float* __restrict__ energies,  // (B, T)
           float* __restrict__ weights)         // (B, T)
{
    __shared__ float redA[8];
    __shared__ float redB[8];
    const int b   = blockIdx.x;
    const int tid = threadIdx.x;                  // 256 threads, 8 vals each
    const float* __restrict__ e = energies + (size_t)b * T_;

    float vals[8];
    float m = -INFINITY;
#pragma unroll
    for (int j = 0; j < 8; ++j) {
        vals[j] = e[tid + j * 256];
        m = fmaxf(m, vals[j]);
    }
#pragma unroll
    for (int off = 16; off > 0; off >>= 1)
        m = fmaxf(m, __shfl_xor(m, off, 32));
    if ((tid & 31) == 0) redA[tid >> 5] = m;
    __syncthreads();
    float bm = redA[0];
#pragma unroll
    for (int i = 1; i < 8; ++i) bm = fmaxf(bm, redA[i]);

    float s = 0.f;
#pragma unroll
    for (int j = 0; j < 8; ++j) {
        vals[j] = expf(vals[j] - bm);
        s += vals[j];
    }
#pragma unroll
    for (int off = 16; off > 0; off >>= 1)
        s += __shfl_xor(s, off, 32);
    if ((tid & 31) == 0) redB[tid >> 5] = s;
    __syncthreads();
    float tot = 0.f;
#pragma unroll
    for (int i = 0; i < 8; ++i) tot += redB[i];
    const float inv = 1.f / tot;

    float* __restrict__ w = weights + (size_t)b * T_;
#pragma unroll
    for (int j = 0; j < 8; ++j)
        w[tid + j * 256] = vals[j] * inv;
}

// ---------------------------------------------------------------------------
// K3: context[b, d] = sum_t w[b,t] * memory[b,t,d]; streams `memory` once.
// Grid (EMB/128, B), 256 threads: 128 d-columns, 2 t-rows in flight.
// ---------------------------------------------------------------------------
__global__ void __launch_bounds__(256)
k3_context(const float* __restrict__ weights,  // (B, T)
           const float* __restrict__ memory,   // (B, T, EMB_DIM)
           float* __restrict__ context)        // (B, EMB_DIM)
{
    __shared__ float sw[T_];                     // 8 KB
    __shared__ float sred[256];
    const int b   = blockIdx.y;
    const int tid = threadIdx.x;
    const int d   = blockIdx.x * 128 + (tid & 127);
    const int tl  = tid >> 7;

    for (int i = tid; i < T_; i += 256)
        sw[i] = weights[(size_t)b * T_ + i];
    __syncthreads();

    const float* __restrict__ mb = memory + (size_t)b * T_ * EMB_DIM;
    float acc = 0.f;
    for (int t = tl; t < T_; t += 2)
        acc = fmaf(sw[t],
                   __builtin_nontemporal_load(&mb[(size_t)t * EMB_DIM + d]),
                   acc);
    sred[tid] = acc;
    __syncthreads();
    if (tid < 128)
        context[(size_t)b * EMB_DIM + d] = sred[tid] + sred[tid + 128];
}

// ---------------------------------------------------------------------------
extern "C" void kernel_launch(void* const* d_in, const int* in_sizes, int n_in,
                              void* d_out, int out_size, void* d_ws, size_t ws_size,
                              hipStream_t stream) {
    const float* hidden = (const float*)d_in[0];  // (B, RNN_DIM)
    const float* memory = (const float*)d_in[1];  // (B, T, EMB_DIM)
    const float* pm     = (const float*)d_in[2];  // (B, T, ATT_DIM)
    const float* cat    = (const float*)d_in[3];  // (B, 2, T)
    /* d_in[4] = mask: unused by reference */
    const float* Wq     = (const float*)d_in[5];  // (ATT_DIM, RNN_DIM)
    const float* convw  = (const float*)d_in[6];  // (N_FILT, 2, KSIZE)
    const float* Wp     = (const float*)d_in[7];  // (ATT_DIM, N_FILT)
    const float* Wv     = (const float*)d_in[8];  // (1, ATT_DIM)

    float* out     = (float*)d_out;
    float* outCtx  = out;                         // (B, EMB_DIM)
    float* outW    = out + B_ * EMB_DIM;          // (B, T)

    float* pq       = (float*)d_ws;               // (B, ATT_DIM)
    float* energies = pq + B_ * ATT_DIM;          // (B, T)

    k0_pq      <<<dim3(B_),           dim3(ATT_DIM), 0, stream>>>(hidden, Wq, pq);
    k1_energies<<<dim3(T_ / TT, B_),  dim3(256),     0, stream>>>(cat, pm, convw, Wp, Wv, pq, energies);
    k2_softmax <<<dim3(B_),           dim3(256),     0, stream>>>(energies, outW);
    k3_context <<<dim3(EMB_DIM/128, B_), dim3(256),  0, stream>>>(outW, memory, outCtx);
}